// NonSymmetricContraction_34668976013687
// MI455X (gfx1250) — compile-verified
//
#include <hip/hip_runtime.h>

typedef __attribute__((ext_vector_type(2))) float v2f;
typedef __attribute__((ext_vector_type(4))) float f4v;
typedef __attribute__((ext_vector_type(8))) float v8f;

#define MUL    128
#define DIMI   4
#define NT     16          // samples per workgroup
#define YSTR   17          // padded sample stride (17 coprime to 64 banks)
#define N_C1   16
#define N_C2   64
#define N_C3   256
#define CU_TOT (N_C1 + N_C2 + N_C3)     // 336
#define CU_FLOATS (CU_TOT * MUL)        // 43008 (172032 B, 256B multiple)
#define SCALE  0.088388347648318447f    // 1/sqrt(128)

// ---------------------------------------------------------------------------
// Kernel 1: precompute Cu[idx][u] = sum_w C[idx, w] * wN[w, u]
// idx 0..15   : C1 (a*4+i)          , 2 paths
// idx 16..79  : C2 (a*16+b*4+i)     , 5 paths
// idx 80..335 : C3 (a*64+b*16+c*4+i), 12 paths
// ---------------------------------------------------------------------------
__global__ void cu_precompute(const float* __restrict__ C1,
                              const float* __restrict__ C2,
                              const float* __restrict__ C3,
                              const float* __restrict__ w1,
                              const float* __restrict__ w2,
                              const float* __restrict__ w3,
                              float* __restrict__ Cu) {
  int g = blockIdx.x * blockDim.x + threadIdx.x;
  if (g >= CU_FLOATS) return;
  int idx = g >> 7;
  int u   = g & (MUL - 1);
  float s = 0.f;
  if (idx < N_C1) {
    for (int w = 0; w < 2; ++w)  s += C1[idx * 2 + w] * w1[w * MUL + u];
  } else if (idx < N_C1 + N_C2) {
    int e = idx - N_C1;
    for (int w = 0; w < 5; ++w)  s += C2[e * 5 + w] * w2[w * MUL + u];
  } else {
    int e = idx - N_C1 - N_C2;
    for (int w = 0; w < 12; ++w) s += C3[e * 12 + w] * w3[w * MUL + u];
  }
  Cu[idx * MUL + u] = s;
}

// ---------------------------------------------------------------------------
// Kernel 1b: repack weights into WMMA-B friendly layout.
// Wp[dblk][k][hl][v][p] = W[dblk][4k + 2*hl + p][v]
// so a B-operand lane fetches its (K, K+1) pair with ONE b64 load, and the
// 16 lanes of a half-wave read one contiguous 128B segment.
// ---------------------------------------------------------------------------
__global__ void w_repack(const float* __restrict__ lin_w,
                         float* __restrict__ Wp) {
  int g = blockIdx.x * blockDim.x + threadIdx.x;   // 0 .. 6*128*128-1
  if (g >= 6 * MUL * MUL) return;
  int p    = g & 1;
  int v    = (g >> 1) & 127;
  int hl   = (g >> 9) & 1;
  int k    = (g >> 10) & 31;
  int dblk = g >> 15;
  Wp[g] = lin_w[(size_t)dblk * MUL * MUL + (4 * k + 2 * hl + p) * MUL + v];
}

// ---------------------------------------------------------------------------
// Kernel 2: fused linears (f32 WMMA) + per-(n,u) contraction
// ---------------------------------------------------------------------------
__global__ __launch_bounds__(256) void nsc_main(
    const float* __restrict__ x,      // (B, 4*128)
    const float* __restrict__ Wp,     // repacked (6,32,2,128,2)
    const float* __restrict__ Cu,     // (336, 128) precomputed
    float* __restrict__ out) {        // (B, 4*128)

  // y[deg][d][v][sample] with padded sample stride
  __shared__ float lds_y[12 * MUL * YSTR];                  // 104448 B
  __shared__ float lds_cu[(N_C1 + N_C2) * MUL];             //  40960 B

  const int tid  = threadIdx.x;
  const int lane = tid & 31;
  const int wv   = tid >> 5;
  const int n0   = blockIdx.x * NT;

  // --- stage C1u / C2u into LDS (b128 coalesced) ---
  for (int e = tid; e < (N_C1 + N_C2) * MUL / 4; e += 256)
    ((f4v*)lds_cu)[e] = ((const f4v*)Cu)[e];

  // =========================================================================
  // Phase A: Y[deg] = X @ W[deg][blk(d)] * (1/sqrt(128)) via f32 WMMA.
  // 12 slabs x 2 v-tile halves = 24 equal units; wave w takes units
  // {w, w+8, w+16} -> perfectly balanced (3 units = 384 WMMAs each).
  // A slab (16x128 f32) lives in registers; per unit: 4 v-tiles x 32 WMMA.
  // =========================================================================
  const int hl = lane >> 4;        // half-wave select (K pairs / M+8)
  const int ll = lane & 15;        // row / column within tile

  for (int q = wv; q < 24; q += 8) {            // wave-uniform
    const int t    = q >> 1;                    // slab 0..11  (deg*4 + d)
    const int h    = q & 1;                     // v-tile half
    const int deg  = t >> 2;
    const int d    = t & 3;
    const int dblk = deg * 2 + ((d == 0) ? 0 : 1);

    // A operand: lane ll -> sample row, hl selects K={0,1} vs K={2,3}
    const float* xr = x + (size_t)(n0 + ll) * (DIMI * MUL) + d * MUL + (hl << 1);
    v2f a[32];
#pragma unroll
    for (int k = 0; k < 32; ++k)
      a[k] = *(const v2f*)(xr + 4 * k);

    const float* wb = Wp + (((size_t)dblk * 32) * 2 + hl) * 256 + ll * 2;

    for (int vt = h * 4; vt < h * 4 + 4; ++vt) {
      v8f c = {0.f, 0.f, 0.f, 0.f, 0.f, 0.f, 0.f, 0.f};
#pragma unroll
      for (int k = 0; k < 32; ++k) {
        // one b64 load per WMMA: (W[4k+2hl][v], W[4k+2hl+1][v])
        v2f b = *(const v2f*)(wb + (size_t)k * 1024 + vt * 32);
        c = __builtin_amdgcn_wmma_f32_16x16x4_f32(
                false, a[k], false, b, (short)0, c, false, false);
      }
      // D tile: lane ll = column v, VGPR j -> sample = j + 8*hl
      const int v = vt * 16 + ll;
      float* yp = &lds_y[((size_t)t * MUL + v) * YSTR + (hl << 3)];
#pragma unroll
      for (int j = 0; j < 8; ++j)
        yp[j] = c[j] * SCALE;
    }
  }

  __syncthreads();

  // =========================================================================
  // Phase B: out[n,i,u] = deg1 + deg2 + deg3 contractions.
  // lane -> u offset (stride-1 banked LDS), wave -> (u chunk, sample half).
  // =========================================================================
  const int u  = ((wv & 3) << 5) + lane;   // 0..127
  const int sb = (wv >> 2) << 3;           // 0 or 8 (8 samples per wave)

  // hoist y for my 8 samples into registers
  float y1r[4][8], y2r[4][8], y3r[4][8];
#pragma unroll
  for (int dd = 0; dd < 4; ++dd) {
#pragma unroll
    for (int s = 0; s < 8; ++s) {
      y1r[dd][s] = lds_y[((size_t)(0 * 4 + dd) * MUL + u) * YSTR + sb + s];
      y2r[dd][s] = lds_y[((size_t)(1 * 4 + dd) * MUL + u) * YSTR + sb + s];
      y3r[dd][s] = lds_y[((size_t)(2 * 4 + dd) * MUL + u) * YSTR + sb + s];
    }
  }

  float acc[8][4];
#pragma unroll
  for (int s = 0; s < 8; ++s)
#pragma unroll
    for (int i = 0; i < 4; ++i) acc[s][i] = 0.f;

  // ---- degree 1: out_i += sum_a C1u[a,i] * y1[a] ----
  {
    float c1[16];
#pragma unroll
    for (int e = 0; e < 16; ++e) c1[e] = lds_cu[e * MUL + u];
#pragma unroll
    for (int s = 0; s < 8; ++s)
#pragma unroll
      for (int a = 0; a < 4; ++a)
#pragma unroll
        for (int i = 0; i < 4; ++i)
          acc[s][i] += c1[a * 4 + i] * y1r[a][s];
  }

  // ---- degree 2: out_i += sum_b y2[b] * (sum_a C2u[a,b,i] * y1[a]) ----
#pragma unroll
  for (int b = 0; b < 4; ++b) {
    float c2[16];
#pragma unroll
    for (int a = 0; a < 4; ++a)
#pragma unroll
      for (int i = 0; i < 4; ++i)
        c2[a * 4 + i] = lds_cu[(N_C1 + a * 16 + b * 4 + i) * MUL + u];
#pragma unroll
    for (int s = 0; s < 8; ++s) {
      float t0 = 0.f, t1 = 0.f, t2 = 0.f, t3 = 0.f;
#pragma unroll
      for (int a = 0; a < 4; ++a) {
        t0 += c2[a * 4 + 0] * y1r[a][s];
        t1 += c2[a * 4 + 1] * y1r[a][s];
        t2 += c2[a * 4 + 2] * y1r[a][s];
        t3 += c2[a * 4 + 3] * y1r[a][s];
      }
      const float yb = y2r[b][s];
      acc[s][0] += yb * t0;  acc[s][1] += yb * t1;
      acc[s][2] += yb * t2;  acc[s][3] += yb * t3;
    }
  }

  // ---- degree 3: out_i += sum_{a,b} y1[a]y2[b] * (sum_c C3u[a,b,c,i]*y3[c])
  const float* C3u = Cu + (size_t)(N_C1 + N_C2) * MUL;
#pragma unroll
  for (int a = 0; a < 4; ++a) {
#pragma unroll
    for (int b = 0; b < 4; ++b) {
      float c3[16];
#pragma unroll
      for (int cc = 0; cc < 4; ++cc)
#pragma unroll
        for (int i = 0; i < 4; ++i)
          c3[cc * 4 + i] = C3u[(size_t)(a * 64 + b * 16 + cc * 4 + i) * MUL + u];
#pragma unroll
      for (int s = 0; s < 8; ++s) {
        const float g = y1r[a][s] * y2r[b][s];
        float t0 = 0.f, t1 = 0.f, t2 = 0.f, t3 = 0.f;
#pragma unroll
        for (int cc = 0; cc < 4; ++cc) {
          t0 += c3[cc * 4 + 0] * y3r[cc][s];
          t1 += c3[cc * 4 + 1] * y3r[cc][s];
          t2 += c3[cc * 4 + 2] * y3r[cc][s];
          t3 += c3[cc * 4 + 3] * y3r[cc][s];
        }
        acc[s][0] += g * t0;  acc[s][1] += g * t1;
        acc[s][2] += g * t2;  acc[s][3] += g * t3;
      }
    }
  }

  // ---- store (coalesced 128B per (n,i)) ----
#pragma unroll
  for (int s = 0; s < 8; ++s) {
    const size_t base = (size_t)(n0 + sb + s) * (DIMI * MUL) + u;
#pragma unroll
    for (int i = 0; i < 4; ++i)
      out[base + i * MUL] = acc[s][i];
  }
}

// ---------------------------------------------------------------------------
extern "C" void kernel_launch(void* const* d_in, const int* in_sizes, int n_in,
                              void* d_out, int out_size, void* d_ws, size_t ws_size,
                              hipStream_t stream) {
  const float* x     = (const float*)d_in[0];
  const float* lin_w = (const float*)d_in[1];
  const float* w1    = (const float*)d_in[2];
  const float* w2    = (const float*)d_in[3];
  const float* w3    = (const float*)d_in[4];
  const float* C1    = (const float*)d_in[5];
  const float* C2    = (const float*)d_in[6];
  const float* C3    = (const float*)d_in[7];
  float* out = (float*)d_out;

  float* Cu = (float*)d_ws;                 // 43008 floats (172032 B)
  float* Wp = Cu + CU_FLOATS;               // 98304 floats (393216 B)

  const int n = in_sizes[0] / (DIMI * MUL);

  cu_precompute<<<(CU_FLOATS + 255) / 256, 256, 0, stream>>>(
      C1, C2, C3, w1, w2, w3, Cu);
  w_repack<<<(6 * MUL * MUL + 255) / 256, 256, 0, stream>>>(lin_w, Wp);
  nsc_main<<<n / NT, 256, 0, stream>>>(x, Wp, Cu, out);
}